// graph_constructor_61564061221147
// MI455X (gfx1250) — compile-verified
//
#include <hip/hip_runtime.h>
#include <hip/hip_bf16.h>

#define DIM 40
#define KDIM 80          // fused K (2*DIM)
#define KSEL 32          // top-k
#define ALPHA 3.0f
#define TILE 64          // workgroup output tile (64x64)
#define LDSS 82          // LDS row stride in floats (even -> 8B aligned pairs; 82 mod 64 = 18 -> low conflict)

typedef __attribute__((ext_vector_type(2))) float v2f;
typedef __attribute__((ext_vector_type(8))) float v8f;

// ---------------------------------------------------------------------------
// Kernel 1: gather + two 40x40 linears + tanh, pack fused operands:
//   U[n] = [tanh(3(W1 e1 + b1)), -tanh(3(W2 e2 + b2))]   (80 floats)
//   V[n] = [tanh(3(W2 e2 + b2)),  tanh(3(W1 e1 + b1))]   (80 floats)
// Rows n >= N are zero padding (so edge tiles compute 0).
// ---------------------------------------------------------------------------
__global__ __launch_bounds__(256) void prep_kernel(
    const int* __restrict__ idx,
    const float* __restrict__ emb1, const float* __restrict__ emb2,
    const float* __restrict__ W1, const float* __restrict__ b1,
    const float* __restrict__ W2, const float* __restrict__ b2,
    float* __restrict__ U, float* __restrict__ V, int n, int npad)
{
    int t = blockIdx.x * 256 + threadIdx.x;
    if (t >= npad * DIM) return;
    int node = t / DIM;
    int d    = t % DIM;
    size_t base = (size_t)node * KDIM;
    if (node >= n) {
        U[base + d] = 0.0f; U[base + DIM + d] = 0.0f;
        V[base + d] = 0.0f; V[base + DIM + d] = 0.0f;
        return;
    }
    int g = idx[node];
    const float* e1 = emb1 + (size_t)g * DIM;
    const float* e2 = emb2 + (size_t)g * DIM;
    const float* w1 = W1 + (size_t)d * DIM;
    const float* w2 = W2 + (size_t)d * DIM;
    float s1 = b1[d], s2 = b2[d];
#pragma unroll 8
    for (int k = 0; k < DIM; ++k) {
        s1 += e1[k] * w1[k];
        s2 += e2[k] * w2[k];
    }
    float y1 = tanhf(ALPHA * s1);
    float y2 = tanhf(ALPHA * s2);
    U[base + d] = y1;  U[base + DIM + d] = -y2;
    V[base + d] = y2;  V[base + DIM + d] =  y1;
}

// ---------------------------------------------------------------------------
// Kernel 2: fp32 WMMA GEMM  adj[i][j] = relu(tanh(3 * (U_i . V_j)))
// 512 threads = 16 waves; each wave owns one 16x16 tile of a 64x64 block tile.
// K-loop: 20 x V_WMMA_F32_16X16X4_F32 accumulating into v8f.
// ---------------------------------------------------------------------------
__global__ __launch_bounds__(512) void gemm_kernel(
    const float* __restrict__ U, const float* __restrict__ V,
    float* __restrict__ out, int n)
{
    __shared__ float As[TILE * LDSS];
    __shared__ float Bs[TILE * LDSS];

    const int rowBase = blockIdx.y * TILE;
    const int colBase = blockIdx.x * TILE;
    const int tid = threadIdx.x;

    // cooperative panel load: 64 rows x 80 floats each for A and B (float4 global)
    for (int i = tid; i < TILE * (KDIM / 4); i += 512) {
        int r = i / (KDIM / 4);
        int c = (i % (KDIM / 4)) * 4;
        float4 a = *(const float4*)(U + (size_t)(rowBase + r) * KDIM + c);
        float4 b = *(const float4*)(V + (size_t)(colBase + r) * KDIM + c);
        float* pa = &As[r * LDSS + c];
        pa[0] = a.x; pa[1] = a.y; pa[2] = a.z; pa[3] = a.w;
        float* pb = &Bs[r * LDSS + c];
        pb[0] = b.x; pb[1] = b.y; pb[2] = b.z; pb[3] = b.w;
    }
    __syncthreads();

    const int lane = tid & 31;
    const int wave = tid >> 5;        // 0..15
    const int wm = wave >> 2;         // 0..3 (row of 16-tiles)
    const int wn = wave & 3;          // 0..3 (col of 16-tiles)
    const int l16  = lane & 15;
    const int half = lane >> 4;       // 0 or 1
    const int koff = 2 * half;        // ISA 16x4 f32 A/B fragment layout

    const float* aRow = &As[(wm * 16 + l16) * LDSS];
    const float* bRow = &Bs[(wn * 16 + l16) * LDSS];

    v8f acc = {0.0f, 0.0f, 0.0f, 0.0f, 0.0f, 0.0f, 0.0f, 0.0f};
#pragma unroll
    for (int kk = 0; kk < KDIM; kk += 4) {
        v2f a = *(const v2f*)(aRow + kk + koff);
        v2f b = *(const v2f*)(bRow + kk + koff);
        acc = __builtin_amdgcn_wmma_f32_16x16x4_f32(
            false, a, false, b, (short)0, acc, false, false);
    }

    // store: VGPR g -> row (wm*16 + 8*half + g), col (wn*16 + l16)
    const int col  = colBase + wn * 16 + l16;
    const int rowb = rowBase + wm * 16 + 8 * half;
    if (col < n) {
#pragma unroll
        for (int g = 0; g < 8; ++g) {
            int row = rowb + g;
            if (row < n) {
                float c = acc[g];
                out[(size_t)row * n + col] = fmaxf(tanhf(ALPHA * c), 0.0f);
            }
        }
    }
}

// ---------------------------------------------------------------------------
// Kernel 3: per-row top-K mask, in place. One 256-thread block per row.
// Values are >= 0 (relu), so float bit patterns order identically to values.
// 4-level radix select (8 bits/level) finds the K-th largest key T, then
// elements > T are kept and the first (K - count(>T)) elements == T in index
// order are kept (matches jax.lax.top_k tie-breaking). Everything else -> 0.
// ---------------------------------------------------------------------------
__global__ __launch_bounds__(256) void topk_kernel(float* __restrict__ out, int n)
{
    __shared__ float rowv[10000];
    __shared__ unsigned int hist[256];
    __shared__ unsigned int eqcnt[256];
    __shared__ unsigned int sh_bin, sh_rem;

    const int tid = threadIdx.x;
    float* src = out + (size_t)blockIdx.x * n;

    for (int i = tid; i < n; i += 256) rowv[i] = src[i];
    __syncthreads();

    unsigned int prefix = 0;
    unsigned int rem = KSEL;
#pragma unroll
    for (int level = 0; level < 4; ++level) {
        const int shift = 24 - 8 * level;
        hist[tid] = 0;
        __syncthreads();
        for (int i = tid; i < n; i += 256) {
            unsigned int key = __float_as_uint(rowv[i]);
            if (level == 0 || (key >> (shift + 8)) == prefix)
                atomicAdd(&hist[(key >> shift) & 255u], 1u);
        }
        __syncthreads();
        if (tid == 0) {
            unsigned int cum = 0;
            int b = 255;
            while (b > 0) {
                unsigned int h = hist[b];
                if (cum + h >= rem) break;
                cum += h;
                --b;
            }
            sh_bin = (unsigned int)b;
            sh_rem = rem - cum;
        }
        __syncthreads();
        prefix = (prefix << 8) | sh_bin;
        rem = sh_rem;
        __syncthreads();
    }
    const unsigned int T = prefix;   // key of K-th largest
    // rem = number of T-valued elements to keep (first ones by index)

    // exact per-thread rank among equal elements via contiguous chunks
    const int chunk = (n + 255) / 256;
    const int beg = tid * chunk;
    const int end = (beg + chunk < n) ? (beg + chunk) : n;
    unsigned int cnt = 0;
    for (int i = beg; i < end; ++i)
        if (__float_as_uint(rowv[i]) == T) ++cnt;
    eqcnt[tid] = cnt;
    __syncthreads();
    unsigned int rank = 0;
    for (int t = 0; t < tid; ++t) rank += eqcnt[t];

    for (int i = beg; i < end; ++i) {
        float v = rowv[i];
        unsigned int key = __float_as_uint(v);
        bool keep;
        if (key > T) {
            keep = true;
        } else if (key == T) {
            keep = (rank < rem);
            ++rank;
        } else {
            keep = false;
        }
        src[i] = keep ? v : 0.0f;
    }
}

// ---------------------------------------------------------------------------
extern "C" void kernel_launch(void* const* d_in, const int* in_sizes, int n_in,
                              void* d_out, int out_size, void* d_ws, size_t ws_size,
                              hipStream_t stream) {
    const int*   idx  = (const int*)d_in[0];
    const float* emb1 = (const float*)d_in[1];
    const float* emb2 = (const float*)d_in[2];
    const float* W1   = (const float*)d_in[3];
    const float* b1   = (const float*)d_in[4];
    const float* W2   = (const float*)d_in[5];
    const float* b2   = (const float*)d_in[6];
    float* out = (float*)d_out;

    const int n      = in_sizes[0];            // 10000
    const int ntiles = (n + TILE - 1) / TILE;  // 157
    const int npad   = ntiles * TILE;          // 10048

    float* U = (float*)d_ws;                   // npad x 80
    float* V = U + (size_t)npad * KDIM;        // npad x 80  (~6.4 MB total)

    int prepThreads = npad * DIM;
    prep_kernel<<<(prepThreads + 255) / 256, 256, 0, stream>>>(
        idx, emb1, emb2, W1, b1, W2, b2, U, V, n, npad);

    gemm_kernel<<<dim3(ntiles, ntiles), 512, 0, stream>>>(U, V, out, n);

    topk_kernel<<<n, 256, 0, stream>>>(out, n);
}